// SKA_60387240182286
// MI455X (gfx1250) — compile-verified
//
#include <hip/hip_runtime.h>

// SKA-style per-pixel dynamic depthwise conv, fp32, banded-GEMM on
// V_WMMA_F32_16X16X4_F32 (CDNA5 / gfx1250, wave32) with async Global->LDS
// staging of the activation slab.
//
// x:[8,256,64,64], w:[8,8,25,64,64], out:[8,256,64,64]
// out[b,c,h,w] = sum_{di,dj} x[b,c,h+di-2,w+dj-2] * w[b,g(c),di*5+dj,h,w]

typedef float v2f __attribute__((ext_vector_type(2)));
typedef float v8f __attribute__((ext_vector_type(8)));
typedef int   v2i __attribute__((ext_vector_type(2)));

#define B_   8
#define C_   256
#define H_   64
#define W_   64
#define G_   8
#define CG   32
#define KS   5
#define K2   25
#define PAD  2
#define HW   (H_ * W_)

#define XROWS  5                 // h-2 .. h+2
#define XCOLS  68                // -2 .. 65 (zero padded)
#define XCH    32                // channels per group
#define XCHSTR (XROWS * XCOLS)   // 340 floats per channel
#define BSTR   20                // bexp m-stride: 20 input columns per 16-col strip

#if defined(__gfx1250__) && \
    __has_builtin(__builtin_amdgcn_global_load_async_to_lds_b64) && \
    __has_builtin(__builtin_amdgcn_s_wait_asynccnt)
#define USE_ASYNC 1
typedef __attribute__((address_space(1))) v2i g_v2i;   // global (device) AS payload ptr
typedef __attribute__((address_space(3))) v2i l_v2i;   // LDS (shared) AS payload ptr
#else
#define USE_ASYNC 0
#endif

__global__ __launch_bounds__(256) void ska_wmma_kernel(
    const float* __restrict__ x, const float* __restrict__ w,
    float* __restrict__ out)
{
    // x slab:  [ch][row][col], 43.5 KB
    __shared__ float xlds[XCH * XCHSTR];
    // expanded banded weights: [di][tile][j][m], 25.6 KB
    //   bexp[di][tile][j][m] = w[k=di*5+(m-j)][tile*16+j] if 0<=m-j<=4 else 0
    __shared__ float bexp[KS * 4 * 16 * BSTR];

    const int bIdx = blockIdx.x;            // B*G*H blocks
    const int h  = bIdx % H_;
    const int g  = (bIdx / H_) % G_;
    const int bb = bIdx / (H_ * G_);
    const int tid = threadIdx.x;

    const int cbase = g * CG;
    const float* xb = x + (size_t)bb * C_ * HW;
    const float* wb = w + (((size_t)bb * G_ + g) * K2) * HW + h * W_;

    // ---- halo columns (0,1 and 66,67) of every (ch,row): zeros
    for (int i = tid; i < XCH * XROWS; i += 256) {
        int ch = i / XROWS, row = i - (i / XROWS) * XROWS;
        v2f z = {};
        *reinterpret_cast<v2f*>(&xlds[ch * XCHSTR + row * XCOLS + 0])  = z;
        *reinterpret_cast<v2f*>(&xlds[ch * XCHSTR + row * XCOLS + 66]) = z;
    }
    // ---- interior cols 0..63 -> LDS cols 2..65, rows clamped to zero.
    // 32 b64 units per (ch,row); in-range rows go through the async
    // Global->LDS path (ASYNCcnt), out-of-range rows are zero-filled.
    for (int i = tid; i < XCH * XROWS * 32; i += 256) {
        int ch  = i / (XROWS * 32);
        int rem = i - ch * (XROWS * 32);
        int row = rem >> 5;
        int cu  = rem & 31;                 // column unit: 2 floats
        int r   = h + row - PAD;
        float* lp = &xlds[ch * XCHSTR + row * XCOLS + 2 + 2 * cu];
        if (r >= 0 && r < H_) {
            const float* gp = xb + (size_t)(cbase + ch) * HW + r * W_ + 2 * cu;
#if USE_ASYNC
            __builtin_amdgcn_global_load_async_to_lds_b64(
                (g_v2i*)gp, (l_v2i*)lp, 0, 0);
#else
            *reinterpret_cast<v2f*>(lp) = *reinterpret_cast<const v2f*>(gp);
#endif
        } else {
            v2f z = {};
            *reinterpret_cast<v2f*>(lp) = z;
        }
    }
    // ---- zero the expanded band matrix
    for (int i = tid; i < KS * 4 * 16 * BSTR; i += 256) bexp[i] = 0.0f;
    __syncthreads();   // bexp zeros visible before scatter

    // ---- scatter the 25x64 weight plane into the band (each elem lands once)
    for (int i = tid; i < K2 * W_; i += 256) {
        int k    = i >> 6;                  // 0..24
        int col  = i & 63;
        int di   = k / KS;
        int dd   = k - di * KS;
        int tile = col >> 4;
        int j    = col & 15;
        bexp[((di * 4 + tile) * 16 + j) * BSTR + (j + dd)] =
            __builtin_nontemporal_load(&wb[(size_t)k * HW + col]);
    }
#if USE_ASYNC
    __builtin_amdgcn_s_wait_asynccnt(0);    // our async LDS writes landed
#endif
    __syncthreads();

    // ---- 8 waves: (column strip tile 0..3) x (channel half 0..1)
    const int lane  = tid & 31;
    const int wid   = tid >> 5;
    const int halfc = wid & 1;
    const int tile  = wid >> 1;
    const int w0    = tile * 16;

    const int j  = lane & 15;               // N (out column) == M (channel) idx
    const int hs = lane >> 4;               // K-pair selector: {0,1} vs {2,3}

    v8f acc = {};

    // A (16x4 f32): lane M=j, VGPR0=K{0|2}, VGPR1=K{1|3}  -> one ds_load_b64
    // B (4x16 f32): VGPR0 rows K{0|2}, VGPR1 rows K{1|3}  -> one ds_load_b64
    // All (di,t) deltas fold into immediate DS offsets; bank-conflict-free.
    const float* abase = &xlds[(halfc * 16 + j) * XCHSTR + w0 + 2 * hs];
    const float* bbase = &bexp[(tile * 16 + j) * BSTR + 2 * hs];
#pragma unroll
    for (int di = 0; di < KS; ++di) {
#pragma unroll
        for (int t = 0; t < 5; ++t) {       // K chunks of 4 over m = 4t..4t+3
            v2f a = *reinterpret_cast<const v2f*>(abase + di * XCOLS + 4 * t);
            v2f b = *reinterpret_cast<const v2f*>(bbase + di * (4 * 16 * BSTR) + 4 * t);
            acc = __builtin_amdgcn_wmma_f32_16x16x4_f32(
                false, a, false, b, (short)0, acc, false, false);
        }
    }

    // ---- store D: VGPR i -> M = i + 8*hs, N = j (streamed, never re-read)
    const int c0 = g * CG + halfc * 16;
    float* ob = out + (size_t)bb * C_ * HW + (size_t)h * W_ + w0 + j;
#pragma unroll
    for (int i = 0; i < 8; ++i) {
        __builtin_nontemporal_store(acc[i], &ob[(size_t)(c0 + i + 8 * hs) * HW]);
    }
}

extern "C" void kernel_launch(void* const* d_in, const int* in_sizes, int n_in,
                              void* d_out, int out_size, void* d_ws, size_t ws_size,
                              hipStream_t stream) {
    const float* x = (const float*)d_in[0];
    const float* w = (const float*)d_in[1];
    float* out = (float*)d_out;
    (void)in_sizes; (void)n_in; (void)out_size; (void)d_ws; (void)ws_size;
    ska_wmma_kernel<<<dim3(B_ * G_ * H_), 256, 0, stream>>>(x, w, out);
}